// CARAFE_2697239462323
// MI455X (gfx1250) — compile-verified
//
#include <hip/hip_runtime.h>

typedef __attribute__((ext_vector_type(2))) float v2f;
typedef __attribute__((ext_vector_type(8))) float v8f;

#define NB   8
#define CIN  256
#define CMID 64
#define CENC 100
#define HH   64
#define WW   64
#define PIX  (HH * WW)
#define HO   128
#define WO   128

// Low 32 bits of a flat __shared__ address are the LDS byte offset
// (CDNA5 ISA 10.2).
#define LDS_OFF(p) ((unsigned)(unsigned long long)(p))

__device__ __forceinline__ void async_ld_lds_b128(unsigned lds_off, const void* gaddr) {
  asm volatile("global_load_async_to_lds_b128 %0, %1, off"
               :: "v"(lds_off), "v"(gaddr) : "memory");
}
__device__ __forceinline__ void async_ld_lds_b32(unsigned lds_off, const void* gaddr) {
  asm volatile("global_load_async_to_lds_b32 %0, %1, off"
               :: "v"(lds_off), "v"(gaddr) : "memory");
}
__device__ __forceinline__ void wait_async0() {
  asm volatile("s_wait_asynccnt 0x0" ::: "memory");
}

// ---------------------------------------------------------------------------
// Kernel 1: 1x1 compress conv as WMMA GEMM.
// comp[n][cm][pix] = sum_k w_comp[cm][k] * x[n][k][pix] + b_comp[cm]
// 8 waves per block, each owning a 16-pixel tile; w_comp (64 KB) is staged
// once per block into LDS with async b128 copies and A-fragments are read
// back as ds_load_b64.
// ---------------------------------------------------------------------------
__global__ __launch_bounds__(256)
void carafe_compress_wmma(const float* __restrict__ x,
                          const float* __restrict__ w_comp,
                          const float* __restrict__ b_comp,
                          float* __restrict__ comp) {
  __shared__ float sW[CMID * CIN];          // 64*256 f32 = 64 KB
  const int n   = blockIdx.y;
  const int tid = threadIdx.x;

  // Stage w_comp: 4096 float4 chunks, 16 per thread, async global->LDS.
#pragma unroll
  for (int i = 0; i < 16; ++i) {
    const int chunk = tid + i * 256;        // float4 index
    async_ld_lds_b128(LDS_OFF(&sW[chunk * 4]), w_comp + chunk * 4);
  }
  wait_async0();
  __syncthreads();

  const int wave    = tid >> 5;
  const int lane    = tid & 31;
  const int pixbase = blockIdx.x * 128 + wave * 16;
  const int nIdx    = lane & 15;            // N (pixel) for B/C/D, M (row) for A
  const int kh      = (lane >> 4) << 1;     // K sub-offset: 0 or 2

  v8f acc[4] = {};
  const float* xb = x + (size_t)n * CIN * PIX + pixbase + nIdx;

  for (int k0 = 0; k0 < CIN; k0 += 4) {
    v2f b;
    b.x = xb[(size_t)(k0 + kh)     * PIX];
    b.y = xb[(size_t)(k0 + kh + 1) * PIX];
#pragma unroll
    for (int mt = 0; mt < 4; ++mt) {
      const int m = mt * 16 + nIdx;
      const v2f a = *(const v2f*)&sW[m * CIN + k0 + kh];   // ds_load_b64
      acc[mt] = __builtin_amdgcn_wmma_f32_16x16x4_f32(
          false, a, false, b, (short)0, acc[mt], false, false);
    }
  }

  const int mrow = (lane >> 4) << 3;        // D: M = v + (lane/16)*8
#pragma unroll
  for (int mt = 0; mt < 4; ++mt) {
#pragma unroll
    for (int v = 0; v < 8; ++v) {
      const int cm = mt * 16 + mrow + v;
      comp[((size_t)n * CMID + cm) * PIX + pixbase + nIdx] = acc[mt][v] + b_comp[cm];
    }
  }
}

// ---------------------------------------------------------------------------
// Kernel 2: 3x3 encoder conv (64 -> 100) as 9 accumulated WMMA GEMMs (K=64).
// Per tap, the 100x64 weight slice (stride-9 gather) is staged into LDS with
// async b32 copies and shared by all 8 waves; rows 100..111 are zero so the
// 7th M-tile's pad rows accumulate zero. Border taps masked per-lane before the
// (fully converged) WMMA.
// ---------------------------------------------------------------------------
__global__ __launch_bounds__(256)
void carafe_encode_wmma(const float* __restrict__ comp,
                        const float* __restrict__ w_enc,
                        const float* __restrict__ b_enc,
                        float* __restrict__ maskbuf) {
  __shared__ float sA[112 * CMID];          // 7168 f32 = 28 KB
  const int n    = blockIdx.y;
  const int tid  = threadIdx.x;
  const int wave = tid >> 5;
  const int lane = tid & 31;

  const int pixbase = blockIdx.x * 128 + wave * 16;
  const int y0   = pixbase >> 6;
  const int x0   = pixbase & 63;
  const int nIdx = lane & 15;
  const int kh   = (lane >> 4) << 1;

  // Zero-fill pad rows 100..111 once (never touched by the async gathers).
#pragma unroll
  for (int j = 0; j < 3; ++j) sA[CENC * CMID + tid * 3 + j] = 0.0f;

  v8f acc[7] = {};
  const float* cbase = comp + (size_t)n * CMID * PIX;

  for (int tap = 0; tap < 9; ++tap) {
    const int ky = tap / 3;                 // 0..2
    const int kx = tap % 3;                 // 0..2

    __syncthreads();                        // prior tap's LDS reads complete
    // Stage A_tap: sA[m*64+k] = w_enc[(m*64+k)*9 + tap]; 6400 elems, 25/thread.
#pragma unroll
    for (int i = 0; i < 25; ++i) {
      const int e = tid + i * 256;
      async_ld_lds_b32(LDS_OFF(&sA[e]), w_enc + (size_t)e * 9 + tap);
    }
    wait_async0();
    __syncthreads();

    const int sy = y0 + ky - 1;
    const int sx = x0 + nIdx + kx - 1;
    const bool valid = (sy >= 0) && (sy < HH) && (sx >= 0) && (sx < WW);
    const int  soff  = sy * WW + sx;

    for (int k0 = 0; k0 < CMID; k0 += 4) {
      const int kk = k0 + kh;
      v2f b;
      b.x = valid ? cbase[(size_t)kk       * PIX + soff] : 0.0f;
      b.y = valid ? cbase[(size_t)(kk + 1) * PIX + soff] : 0.0f;
#pragma unroll
      for (int mt = 0; mt < 7; ++mt) {
        const int m = mt * 16 + nIdx;
        const v2f a = *(const v2f*)&sA[m * CMID + kk];     // ds_load_b64
        acc[mt] = __builtin_amdgcn_wmma_f32_16x16x4_f32(
            false, a, false, b, (short)0, acc[mt], false, false);
      }
    }
  }

  const int mrow = (lane >> 4) << 3;
#pragma unroll
  for (int mt = 0; mt < 7; ++mt) {
#pragma unroll
    for (int v = 0; v < 8; ++v) {
      const int m = mt * 16 + mrow + v;
      if (m < CENC) {
        maskbuf[((size_t)n * CENC + m) * PIX + pixbase + nIdx] =
            acc[mt][v] + b_enc[m];
      }
    }
  }
}

// ---------------------------------------------------------------------------
// Kernel 3: fused pixel-shuffle + softmax(25) + 25-tap reassembly.
// One block per input pixel (n,y,x): 256 threads = 256 channels, each thread
// produces the 2x2 output quad with float2 (b64) stores. Softmax for the 4
// sub-pixel weight vectors is computed once in LDS, shared by all channels.
// Pixel shuffle: conv channel = idx*4 + ry*2 + rx.
// ---------------------------------------------------------------------------
__global__ __launch_bounds__(256)
void carafe_apply(const float* __restrict__ x,
                  const float* __restrict__ maskbuf,
                  float* __restrict__ out) {
  __shared__ float logits[CENC];
  __shared__ float wgt[4][25];

  const int xq  = blockIdx.x;
  const int yq  = blockIdx.y;
  const int n   = blockIdx.z;
  const int tid = threadIdx.x;

  if (tid < CENC) {
    logits[tid] = maskbuf[(((size_t)n * CENC + tid) * HH + yq) * WW + xq];
  }
  __syncthreads();

  if (tid < 4) {
    const int sub = tid;   // ry*2 + rx
    float mx = -1e30f;
#pragma unroll
    for (int i = 0; i < 25; ++i) mx = fmaxf(mx, logits[i * 4 + sub]);
    float s = 0.0f;
    float e[25];
#pragma unroll
    for (int i = 0; i < 25; ++i) {
      e[i] = __expf(logits[i * 4 + sub] - mx);
      s += e[i];
    }
    const float inv = 1.0f / s;
#pragma unroll
    for (int i = 0; i < 25; ++i) wgt[sub][i] = e[i] * inv;
  }
  __syncthreads();

  const int c = tid;
  const float* plane = x + ((size_t)n * CIN + c) * PIX;
  float a00 = 0.f, a01 = 0.f, a10 = 0.f, a11 = 0.f;

#pragma unroll
  for (int dy = 0; dy < 5; ++dy) {
    const int sy = yq + dy - 2;
    const bool vy = (sy >= 0) && (sy < HH);
#pragma unroll
    for (int dx = 0; dx < 5; ++dx) {
      const int sx = xq + dx - 2;
      const bool v = vy && (sx >= 0) && (sx < WW);
      const float val = v ? plane[sy * WW + sx] : 0.0f;
      const int idx = dy * 5 + dx;
      a00 = fmaf(val, wgt[0][idx], a00);
      a01 = fmaf(val, wgt[1][idx], a01);
      a10 = fmaf(val, wgt[2][idx], a10);
      a11 = fmaf(val, wgt[3][idx], a11);
    }
  }

  float* orow0 = out + (((size_t)n * CIN + c) * HO + (yq * 2)) * WO + (xq * 2);
  float* orow1 = orow0 + WO;
  *reinterpret_cast<float2*>(orow0) = make_float2(a00, a01);
  *reinterpret_cast<float2*>(orow1) = make_float2(a10, a11);
}

// ---------------------------------------------------------------------------
extern "C" void kernel_launch(void* const* d_in, const int* in_sizes, int n_in,
                              void* d_out, int out_size, void* d_ws, size_t ws_size,
                              hipStream_t stream) {
  (void)in_sizes; (void)n_in; (void)out_size; (void)ws_size;

  const float* x      = (const float*)d_in[0];   // (8,256,64,64)
  const float* w_comp = (const float*)d_in[1];   // (64,256,1,1)
  const float* b_comp = (const float*)d_in[2];   // (64,)
  const float* w_enc  = (const float*)d_in[3];   // (100,64,3,3)
  const float* b_enc  = (const float*)d_in[4];   // (100,)
  float* out = (float*)d_out;                    // (8,256,128,128)

  float* comp    = (float*)d_ws;                          // 8*64*4096 f32 = 8 MB
  float* maskbuf = comp + (size_t)NB * CMID * PIX;        // 8*100*4096 f32 = 13.1 MB

  carafe_compress_wmma<<<dim3(PIX / 128, NB), 256, 0, stream>>>(x, w_comp, b_comp, comp);
  carafe_encode_wmma  <<<dim3(PIX / 128, NB), 256, 0, stream>>>(comp, w_enc, b_enc, maskbuf);
  carafe_apply        <<<dim3(WW, HH, NB), 256, 0, stream>>>(x, maskbuf, out);
}